// WindowAttention3D_21517786153453
// MI455X (gfx1250) — compile-verified
//
#include <hip/hip_runtime.h>
#include <math.h>

// ---------------------------------------------------------------------------
// WindowAttention3D (focused linear attention) for MI455X / gfx1250 (wave32).
// All GEMMs use V_WMMA_F32_16X16X4_F32 (full fp32 matrix path; the workload
// is bandwidth-bound at 23.3 TB/s so fp32 WMMA costs nothing vs f16).
// This revision eliminates scratch spills: no variably-indexed v8f arrays,
// fixed tiles-per-wave, fully unrolled fragment loops, vectorized A loads.
// ---------------------------------------------------------------------------

typedef __attribute__((ext_vector_type(2))) float v2f;
typedef __attribute__((ext_vector_type(8))) float v8f;

#define CDIM   96
#define HD     32
#define NHEAD  3
#define NTOK   343            // 7*7*7
#define NWIN   512            // 8*8*8
#define BH     (NWIN * NHEAD) // 1536
#define MROWS  175616         // 56*56*56
#define EPSV   1e-6f

__device__ __forceinline__ v8f wmma4(v2f a, v2f b, v8f c) {
  // D = A(16x4,f32) * B(4x16,f32) + C(16x16,f32)
  return __builtin_amdgcn_wmma_f32_16x16x4_f32(false, a, false, b,
                                               (short)0, c, false, false);
}

// flattened (d,h,w) row -> (window index, intra-window token)
__device__ __forceinline__ void m_to_win(int m, int& win, int& n) {
  int d = m / 3136, rem = m % 3136, h = rem / 56, w = rem % 56;
  win = (d / 7) * 64 + (h / 7) * 8 + (w / 7);
  n   = (d % 7) * 49 + (h % 7) * 7 + (w % 7);
}

__device__ __forceinline__ int win_to_m(int win, int n) {
  int wd = win >> 6, wh = (win >> 3) & 7, ww = win & 7;
  int nd = n / 49, r = n % 49, nh = r / 7, nw = r % 7;
  int d = wd * 7 + nd, h = wh * 7 + nh, w = ww * 7 + nw;
  return (d * 56 + h) * 56 + w;
}

// ---------------------------------------------------------------------------
// Kernel 1: qkv = x @ w_qkv, scattered into windowed/head layout
//   q/k/v [bh = win*3+head][n][c]   (bh-major contiguous)
// One block (288 threads = 9 waves) per 16-row strip of M; each wave owns
// exactly 2 of the 18 N-tiles (cols 0..287) in named registers (no spills).
// The A-fragment is a single 64-bit load reused across both N-tiles.
// ---------------------------------------------------------------------------
__global__ __launch_bounds__(288)
void qkv_gemm_scatter(const float* __restrict__ x,
                      const float* __restrict__ wq,
                      float* __restrict__ qb, float* __restrict__ kb,
                      float* __restrict__ vb) {
  const int tid  = threadIdx.x;
  const int lane = tid & 31;
  const int wave = tid >> 5;             // 0..8
  const int mt   = blockIdx.x;           // 0..10975
  const int l15  = lane & 15;
  const int klo  = (lane >> 4) * 2;      // 0 or 2

  const int col0 = wave * 16 + l15;        // N-tile wave
  const int col1 = (wave + 9) * 16 + l15;  // N-tile wave+9

  v8f acc0 = (v8f){};
  v8f acc1 = (v8f){};

  const int row = mt * 16 + l15;         // M row for A fragment
  const float* xrow = x + (size_t)row * CDIM;
  for (int kk = 0; kk < CDIM; kk += 4) {
    const v2f a = *(const v2f*)(xrow + kk + klo);      // 64-bit load
    const float* w0 = wq + (size_t)(kk + klo) * 288;
    v2f b0, b1;
    b0.x = w0[col0];        b0.y = w0[288 + col0];
    b1.x = w0[col1];        b1.y = w0[288 + col1];
    acc0 = wmma4(a, b0, acc0);
    acc1 = wmma4(a, b1, acc1);
  }

  // scatter D tiles into windowed q/k/v (fully unrolled, constant subscripts)
  const int hi8 = 8 * (lane >> 4);
#pragma unroll
  for (int t = 0; t < 2; ++t) {
    const int col   = (t == 0) ? col0 : col1;
    const int which = col / CDIM;        // 0=q 1=k 2=v
    const int cc    = col % CDIM;
    const int head  = cc >> 5;
    const int ch    = cc & 31;
    float* dst = (which == 0) ? qb : (which == 1) ? kb : vb;
#pragma unroll
    for (int r = 0; r < 8; ++r) {
      const int mr = mt * 16 + r + hi8;
      int win, n; m_to_win(mr, win, n);
      const float val = (t == 0) ? acc0[r] : acc1[r];
      dst[((size_t)(win * NHEAD + head) * NTOK + n) * HD + ch] = val;
    }
  }
}

// ---------------------------------------------------------------------------
// Kernel 2: focused-attention feature map on q and k (in place).
// One wave per token row; wave32 shuffle reductions for the L2 norms.
//   k += pos_enc; x = (relu(x)+eps)/softplus(scale); x <- x^3/||x^3||*||x||
// ---------------------------------------------------------------------------
__global__ __launch_bounds__(256)
void qk_transform(float* __restrict__ qb, float* __restrict__ kb,
                  const float* __restrict__ scale_param,
                  const float* __restrict__ pos_enc) {
  const int tid  = threadIdx.x;
  const int lane = tid & 31;
  const int row  = blockIdx.x * 8 + (tid >> 5);   // 0..(BH*NTOK-1)
  const int n    = row % NTOK;

  const float scale = log1pf(expf(scale_param[lane]));  // softplus, per-channel
  const float inv_s = 1.0f / scale;

  const size_t base = (size_t)row * HD + lane;

  // ---- q ----
  {
    float xq = qb[base];
    xq = fmaxf(xq, 0.0f) + EPSV;
    xq *= inv_s;
    float s2 = xq * xq;
#pragma unroll
    for (int m = 16; m >= 1; m >>= 1) s2 += __shfl_xor(s2, m, 32);
    float p  = xq * xq * xq;
    float p2 = p * p;
#pragma unroll
    for (int m = 16; m >= 1; m >>= 1) p2 += __shfl_xor(p2, m, 32);
    qb[base] = p * sqrtf(s2 / p2);
  }
  // ---- k (with positional encoding) ----
  {
    float xk = kb[base] + pos_enc[(size_t)n * HD + lane];
    xk = fmaxf(xk, 0.0f) + EPSV;
    xk *= inv_s;
    float s2 = xk * xk;
#pragma unroll
    for (int m = 16; m >= 1; m >>= 1) s2 += __shfl_xor(s2, m, 32);
    float p  = xk * xk * xk;
    float p2 = p * p;
#pragma unroll
    for (int m = 16; m >= 1; m >>= 1) p2 += __shfl_xor(p2, m, 32);
    kb[base] = p * sqrtf(s2 / p2);
  }
}

// ---------------------------------------------------------------------------
// Kernel 3: linear attention per (window,head) pair.  One block per pair.
//   ksum -> z_i = 1/(q_i . ksum + eps)
//   kv = k^T v   (32x32, four 16x16 WMMA tiles, K loop over 343 tokens)
//   out = (q @ kv) * z    (22 M-strips x 2 N-tiles, WMMA, A-frag shared)
// Output scattered (window-reverse fused) into linear [m][96] buffer.
// ---------------------------------------------------------------------------
__global__ __launch_bounds__(256)
void attention_pair(const float* __restrict__ qb, const float* __restrict__ kb,
                    const float* __restrict__ vb, float* __restrict__ out_lin) {
  __shared__ float ksum[HD];
  __shared__ float zbuf[NTOK];
  __shared__ float kvS[HD * HD];

  const int b    = blockIdx.x;           // 0..1535
  const int tid  = threadIdx.x;
  const int lane = tid & 31;
  const int wave = tid >> 5;
  const int l15  = lane & 15;
  const int klo  = (lane >> 4) * 2;
  const int hi8  = 8 * (lane >> 4);
  const size_t bb = (size_t)b * NTOK * HD;

  // phase 1: ksum[c]
  if (tid < HD) ksum[tid] = 0.0f;
  __syncthreads();
  {
    const int c = lane;
    float part = 0.0f;
    for (int n = wave; n < NTOK; n += 8) part += kb[bb + (size_t)n * HD + c];
    atomicAdd(&ksum[c], part);
  }
  __syncthreads();

  // phase 2: z_i
  for (int n = tid; n < NTOK; n += 256) {
    float dot = 0.0f;
#pragma unroll 8
    for (int c = 0; c < HD; ++c) dot += qb[bb + (size_t)n * HD + c] * ksum[c];
    zbuf[n] = 1.0f / (dot + EPSV);
  }

  // phase 3: kv = k^T v (waves 0..3 own the 2x2 tiles)
  if (wave < 4) {
    const int tm = wave >> 1, tn = wave & 1;
    v8f acc = (v8f){};
    const int crow = tm * 16 + l15;      // c index (A row)
    const int ccol = tn * 16 + l15;      // d index (B col)
    for (int j0 = 0; j0 < NTOK; j0 += 4) {
      const int j1 = j0 + klo, j2 = j0 + klo + 1;
      v2f a, v2;
      a.x  = (j1 < NTOK) ? kb[bb + (size_t)j1 * HD + crow] : 0.0f;
      a.y  = (j2 < NTOK) ? kb[bb + (size_t)j2 * HD + crow] : 0.0f;
      v2.x = (j1 < NTOK) ? vb[bb + (size_t)j1 * HD + ccol] : 0.0f;
      v2.y = (j2 < NTOK) ? vb[bb + (size_t)j2 * HD + ccol] : 0.0f;
      acc = wmma4(a, v2, acc);
    }
#pragma unroll
    for (int r = 0; r < 8; ++r) {
      const int rr = tm * 16 + r + hi8;
      kvS[rr * HD + ccol] = acc[r];
    }
  }
  __syncthreads();

  // phase 4: out = (q @ kv) * z, scattered to linear layout
  const int win  = b / NHEAD;
  const int head = b % NHEAD;
  for (int strip = wave; strip < 22; strip += 8) {
    v8f a0 = (v8f){}, a1 = (v8f){};
    const int na  = strip * 16 + l15;
    const bool va = (na < NTOK);
    const float* qrow = qb + bb + (size_t)(va ? na : 0) * HD;
#pragma unroll
    for (int c0 = 0; c0 < HD; c0 += 4) {
      const int kA = c0 + klo;
      const v2f a = va ? *(const v2f*)(qrow + kA) : (v2f){};  // 64-bit load
      v2f b0, b1;
      b0.x = kvS[kA * HD + l15];       b0.y = kvS[(kA + 1) * HD + l15];
      b1.x = kvS[kA * HD + l15 + 16];  b1.y = kvS[(kA + 1) * HD + l15 + 16];
      a0 = wmma4(a, b0, a0);
      a1 = wmma4(a, b1, a1);
    }
#pragma unroll
    for (int r = 0; r < 8; ++r) {
      const int nr = strip * 16 + r + hi8;
      if (nr < NTOK) {
        const float zz = zbuf[nr];
        const int m = win_to_m(win, nr);
        const int c0 = head * HD + l15;
        out_lin[(size_t)m * CDIM + c0]      = a0[r] * zz;
        out_lin[(size_t)m * CDIM + c0 + 16] = a1[r] * zz;
      }
    }
  }
}

// ---------------------------------------------------------------------------
// Kernel 4: depthwise 5x5x5 conv on v within each window (pad=2), accumulated
// into the linear output buffer.  v tile + weights staged in LDS (~59 KB).
// ---------------------------------------------------------------------------
__global__ __launch_bounds__(256)
void dwconv_add(const float* __restrict__ vb, const float* __restrict__ dwc_w,
                const float* __restrict__ dwc_b, float* __restrict__ out_lin) {
  __shared__ float vS[NTOK * HD];   // 43904 B
  __shared__ float wS[HD * 125];    // 16000 B

  const int b   = blockIdx.x;
  const int tid = threadIdx.x;
  const size_t bb = (size_t)b * NTOK * HD;

  for (int i = tid; i < NTOK * HD; i += 256) vS[i] = vb[bb + i];
  for (int i = tid; i < HD * 125; i += 256)  wS[i] = dwc_w[i];
  __syncthreads();

  const int win  = b / NHEAD;
  const int head = b % NHEAD;
  for (int idx = tid; idx < NTOK * HD; idx += 256) {
    const int n = idx >> 5, c = idx & 31;
    const int od = n / 49, r = n % 49, oh = r / 7, ow = r % 7;
    float acc = dwc_b[c];
    for (int kd = 0; kd < 5; ++kd) {
      const int id = od + kd - 2;
      if ((unsigned)id >= 7u) continue;
#pragma unroll
      for (int kh = 0; kh < 5; ++kh) {
        const int ih = oh + kh - 2;
        if ((unsigned)ih >= 7u) continue;
#pragma unroll
        for (int kw = 0; kw < 5; ++kw) {
          const int iw = ow + kw - 2;
          if ((unsigned)iw >= 7u) continue;
          acc += vS[((id * 49 + ih * 7 + iw) << 5) + c] *
                 wS[c * 125 + kd * 25 + kh * 5 + kw];
        }
      }
    }
    const int m = win_to_m(win, n);
    const size_t o = (size_t)m * CDIM + head * HD + c;
    out_lin[o] += acc;
  }
}

// ---------------------------------------------------------------------------
// Kernel 5: final projection  y = out_lin @ w_proj + b_proj   (96x96)
// One block per 16-row strip; 6 waves = 6 N-tiles of 16.
// ---------------------------------------------------------------------------
__global__ __launch_bounds__(192)
void proj_gemm(const float* __restrict__ out_lin,
               const float* __restrict__ w_proj,
               const float* __restrict__ b_proj,
               float* __restrict__ y) {
  const int tid  = threadIdx.x;
  const int lane = tid & 31;
  const int wave = tid >> 5;            // 0..5 -> N tile
  const int mt   = blockIdx.x;
  const int l15  = lane & 15;
  const int klo  = (lane >> 4) * 2;
  const int hi8  = 8 * (lane >> 4);
  const int col  = wave * 16 + l15;
  const int row  = mt * 16 + l15;

  const float* arow = out_lin + (size_t)row * CDIM;
  v8f acc = (v8f){};
  for (int kk = 0; kk < CDIM; kk += 4) {
    const v2f a = *(const v2f*)(arow + kk + klo);       // 64-bit load
    v2f b;
    b.x = w_proj[(size_t)(kk + klo) * CDIM + col];
    b.y = w_proj[(size_t)(kk + klo + 1) * CDIM + col];
    acc = wmma4(a, b, acc);
  }
  const float bias = b_proj[col];
#pragma unroll
  for (int r = 0; r < 8; ++r) {
    const int mr = mt * 16 + r + hi8;
    y[(size_t)mr * CDIM + col] = acc[r] + bias;
  }
}

// ---------------------------------------------------------------------------
extern "C" void kernel_launch(void* const* d_in, const int* in_sizes, int n_in,
                              void* d_out, int out_size, void* d_ws, size_t ws_size,
                              hipStream_t stream) {
  const float* x           = (const float*)d_in[0];
  const float* w_qkv       = (const float*)d_in[1];
  const float* scale_param = (const float*)d_in[2];
  const float* pos_enc     = (const float*)d_in[3];
  const float* dwc_w       = (const float*)d_in[4];
  const float* dwc_b       = (const float*)d_in[5];
  const float* w_proj      = (const float*)d_in[6];
  const float* b_proj      = (const float*)d_in[7];
  float* y = (float*)d_out;

  const size_t QS = (size_t)BH * NTOK * HD;  // 16,859,136 floats (== MROWS*96)
  float* qb      = (float*)d_ws;
  float* kb      = qb + QS;
  float* vb      = kb + QS;
  float* out_lin = vb + QS;

  // 1) qkv projection + window/head scatter
  qkv_gemm_scatter<<<MROWS / 16, 288, 0, stream>>>(x, w_qkv, qb, kb, vb);
  // 2) focused feature map on q and k
  qk_transform<<<(BH * NTOK) / 8, 256, 0, stream>>>(qb, kb, scale_param, pos_enc);
  // 3) linear attention per (window,head)
  attention_pair<<<BH, 256, 0, stream>>>(qb, kb, vb, out_lin);
  // 4) depthwise conv accumulation
  dwconv_add<<<BH, 256, 0, stream>>>(vb, dwc_w, dwc_b, out_lin);
  // 5) output projection
  proj_gemm<<<MROWS / 16, 192, 0, stream>>>(out_lin, w_proj, b_proj, y);
}